// WindowAdjusted_90108413870739
// MI455X (gfx1250) — compile-verified
//
#include <hip/hip_runtime.h>
#include <hip/hip_bf16.h>

// ---------------------------------------------------------------------------
// Depth-2 path signature prefix scan, closed form:
//   S1(n) = x[n] - x[0]
//   S2(n) = sum_{t<n} a_t b_t^T,  a_t = (x[t]+x[t+1])/2 - x[0],  b_t = x[t+1]-x[t]
// Window w output = [S1(n_w) | S2(n_w)],  n_w = 159 + 32*w,  w = 0..8187.
// Chunk c: increments t in [start_c, 159+32c),  start_0 = 0, start_c = 127+32c.
// G_c = A^T B  (64x64, K=32) via V_WMMA_F32_16X16X4_F32; prefix-sum G over c.
// ---------------------------------------------------------------------------

typedef __attribute__((ext_vector_type(2))) float v2f;
typedef __attribute__((ext_vector_type(8))) float v8f;

#define DIMC   64
#define NW     8188        // number of windows / chunks
#define GROUP  4           // chunks (waves) per workgroup
#define NG     (NW / GROUP) // 2047 groups

// ---------------------------------------------------------------------------
// Phase A: per-chunk 64x64 Gram update via WMMA + in-block prefix over GROUP
// chunks. Writes locally-prefixed S2 rows to out, group totals to ws.
// ---------------------------------------------------------------------------
__global__ __launch_bounds__(32 * GROUP) void sig_chunk_wmma(
    const float* __restrict__ x, float* __restrict__ out,
    float* __restrict__ ws) {
  __shared__ float lds[GROUP][4096];  // 64 KB: per-wave staging, then G slots

  const int wid  = threadIdx.x >> 5;
  const int lane = threadIdx.x & 31;
  const int c    = blockIdx.x * GROUP + wid;  // chunk id, always < NW

  float* slot = &lds[wid][0];  // [0,2048): A pairs, [2048,4096): B pairs

  // Each lane owns column pair (2*lane, 2*lane+1).
  const int   c0  = lane * 2;
  const float x00 = x[c0];
  const float x01 = x[c0 + 1];

  v8f acc[4][4];
  const v8f vzero = {0.f, 0.f, 0.f, 0.f, 0.f, 0.f, 0.f, 0.f};
#pragma unroll
  for (int mi = 0; mi < 4; ++mi)
#pragma unroll
    for (int nj = 0; nj < 4; ++nj) acc[mi][nj] = vzero;

  const int start = (c == 0) ? 0 : (127 + 32 * c);
  const int end   = 159 + 32 * c;            // exclusive increment bound
  const int nsub  = (c == 0) ? 5 : 1;        // chunk 0 carries the 159-inc init

  for (int s = 0; s < nsub; ++s) {
    const int base = start + 32 * s;
    // ---- stage a,b into LDS in fragment-pair layout:
    //      addr = ((t>>1)*64 + col)*2 + (t&1)   (A at 0, B at +2048)
    float cur0 = x[(size_t)base * DIMC + c0];
    float cur1 = x[(size_t)base * DIMC + c0 + 1];
#pragma unroll 4
    for (int t = 0; t < 32; ++t) {
      const size_t r  = (size_t)(base + t + 1) * DIMC;
      const float  n0 = x[r + c0];
      const float  n1 = x[r + c0 + 1];
      float b0 = n0 - cur0, b1 = n1 - cur1;
      float a0 = 0.5f * (cur0 + n0) - x00;
      float a1 = 0.5f * (cur1 + n1) - x01;
      if (base + t >= end) { b0 = 0.f; b1 = 0.f; a0 = 0.f; a1 = 0.f; }  // pad
      const int ia = ((t >> 1) * DIMC + c0) * 2 + (t & 1);
      slot[ia]            = a0;
      slot[ia + 2]        = a1;
      slot[2048 + ia]     = b0;
      slot[2048 + ia + 2] = b1;
      cur0 = n0;
      cur1 = n1;
    }
    __builtin_amdgcn_wave_barrier();  // LDS pipe is in-order per wave; pin sched

    // ---- 8 K-steps x 16 tiles of v_wmma_f32_16x16x4_f32
    const int mcol = lane & 15;
#pragma unroll
    for (int k = 0; k < 8; ++k) {
      const int krow = (2 * k + (lane >> 4)) * (DIMC * 2);
      v2f af[4], bf[4];
#pragma unroll
      for (int mi = 0; mi < 4; ++mi)
        af[mi] = *(const v2f*)&slot[krow + (16 * mi + mcol) * 2];
#pragma unroll
      for (int nj = 0; nj < 4; ++nj)
        bf[nj] = *(const v2f*)&slot[2048 + krow + (16 * nj + mcol) * 2];
#pragma unroll
      for (int mi = 0; mi < 4; ++mi)
#pragma unroll
        for (int nj = 0; nj < 4; ++nj)
          acc[mi][nj] = __builtin_amdgcn_wmma_f32_16x16x4_f32(
              false, af[mi], false, bf[nj], (short)0, acc[mi][nj], false, false);
    }
    __builtin_amdgcn_wave_barrier();
  }

  // ---- dump raw G_c into own slot (row-major m*64+n), then block prefix
  const int half = (lane >> 4) * 8;
  const int ncol = lane & 15;
#pragma unroll
  for (int mi = 0; mi < 4; ++mi)
#pragma unroll
    for (int nj = 0; nj < 4; ++nj)
#pragma unroll
      for (int r = 0; r < 8; ++r)
        slot[(16 * mi + half + r) * DIMC + 16 * nj + ncol] = acc[mi][nj][r];

  __syncthreads();

  for (int j = 0; j < wid; ++j) {  // inclusive prefix within group
    const float* sj = &lds[j][0];
#pragma unroll
    for (int mi = 0; mi < 4; ++mi)
#pragma unroll
      for (int nj = 0; nj < 4; ++nj)
#pragma unroll
        for (int r = 0; r < 8; ++r)
          acc[mi][nj][r] += sj[(16 * mi + half + r) * DIMC + 16 * nj + ncol];
  }

  __syncthreads();

#pragma unroll
  for (int mi = 0; mi < 4; ++mi)  // overwrite own slot with prefixed value
#pragma unroll
    for (int nj = 0; nj < 4; ++nj)
#pragma unroll
      for (int r = 0; r < 8; ++r)
        slot[(16 * mi + half + r) * DIMC + 16 * nj + ncol] = acc[mi][nj][r];

  __syncthreads();

  // ---- coalesced writeout: S2 row for window c; last wave also writes total
  float*       orow = out + (size_t)c * 4160 + 64;   // 16B aligned
  float*       wrow = ws + (size_t)blockIdx.x * 4096;
  const float4* s4  = (const float4*)slot;
  for (int i = lane; i < 1024; i += 32) {
    const float4 v        = s4[i];
    ((float4*)orow)[i]    = v;
    if (wid == GROUP - 1) ((float4*)wrow)[i] = v;
  }
}

// ---------------------------------------------------------------------------
// Phase B: in-place exclusive scan of group totals (NG x 4096).
// ---------------------------------------------------------------------------
__global__ __launch_bounds__(256) void sig_scan_totals(float* __restrict__ ws) {
  const int e = blockIdx.x * blockDim.x + threadIdx.x;  // 0..4095
  float run = 0.f;
  for (int g = 0; g < NG; ++g) {
    const size_t idx = (size_t)g * 4096 + e;
    const float  v   = ws[idx];
    ws[idx]          = run;
    run += v;
  }
}

// ---------------------------------------------------------------------------
// Phase C: add exclusive group offset to each window's S2; write S1.
// ---------------------------------------------------------------------------
__global__ __launch_bounds__(256) void sig_finalize(const float* __restrict__ x,
                                                    float* __restrict__ out,
                                                    const float* __restrict__ ws) {
  const int    w    = blockIdx.x;          // window id, < NW
  const int    g    = w >> 2;              // group id (GROUP == 4)
  const size_t orow = (size_t)w * 4160;
  const float* off  = ws + (size_t)g * 4096;  // L2-resident (34 MB << 192 MB)
  for (int e = threadIdx.x; e < 4096; e += 256)
    out[orow + 64 + e] += off[e];
  if (threadIdx.x < 64) {
    const size_t n = (size_t)(159 + 32 * w) * DIMC;
    out[orow + threadIdx.x] = x[n + threadIdx.x] - x[threadIdx.x];
  }
}

// ---------------------------------------------------------------------------
extern "C" void kernel_launch(void* const* d_in, const int* in_sizes, int n_in,
                              void* d_out, int out_size, void* d_ws,
                              size_t ws_size, hipStream_t stream) {
  const float* x   = (const float*)d_in[0];
  float*       out = (float*)d_out;
  float*       ws  = (float*)d_ws;  // needs NG*4096*4 = 33.5 MB

  hipLaunchKernelGGL(sig_chunk_wmma, dim3(NG), dim3(32 * GROUP), 0, stream, x,
                     out, ws);
  hipLaunchKernelGGL(sig_scan_totals, dim3(4096 / 256), dim3(256), 0, stream,
                     ws);
  hipLaunchKernelGGL(sig_finalize, dim3(NW), dim3(256), 0, stream, x, out, ws);
}